// RUNG_parametric_gamma_81621558493643
// MI455X (gfx1250) — compile-verified
//
#include <hip/hip_runtime.h>
#include <hip/hip_bf16.h>

#define N_NODES 4096
#define IN_DIM  1433
#define HID     64
#define OUT_C   7
#define SCAD_A  3.7f
#define LAM_W   (1.0f / 0.9f - 1.0f)   // lambda = 1/lam_hat - 1
#define K_STEPS 10
#define MAX_DEG 256

typedef __attribute__((ext_vector_type(2))) float v2f;
typedef __attribute__((ext_vector_type(8))) float v8f;

// ---------------------------------------------------------------------------
// Kernel 1: F0 = relu(X @ w1 + b1) @ w2 + b2, one wave per 16-row tile.
// Uses V_WMMA_F32_16X16X4_F32: A is 16x4 f32 (2 VGPRs), B is 4x16 (2 VGPRs),
// C/D 16x16 f32 (8 VGPRs). Four N-tiles of 16 cover HID=64.
// ---------------------------------------------------------------------------
__global__ __launch_bounds__(32) void mlp_f0_wmma(
    const float* __restrict__ X, const float* __restrict__ w1,
    const float* __restrict__ b1, const float* __restrict__ w2,
    const float* __restrict__ b2, float* __restrict__ F0)
{
  __shared__ float hsm[16][HID + 1];

  const int lane  = threadIdx.x;        // 0..31
  const int m0    = blockIdx.x * 16;    // tile base row
  const int mloc  = lane & 15;
  const int khalf = (lane >> 4) * 2;    // 0 for lanes 0-15, 2 for lanes 16-31

  v8f acc[4] = {v8f{}, v8f{}, v8f{}, v8f{}};
  const float* xrow = X + (size_t)(m0 + mloc) * IN_DIM;

  // main loop: k0 in [0, 1432), always fully in-bounds (1432 = 4*358)
  for (int k0 = 0; k0 < IN_DIM - 1; k0 += 4) {
    const int ka = k0 + khalf;
    v2f a;
    a.x = xrow[ka];
    a.y = xrow[ka + 1];
    const float* w1a = w1 + (size_t)ka * HID;
    const float* w1b = w1 + (size_t)(ka + 1) * HID;
#pragma unroll
    for (int t = 0; t < 4; ++t) {
      v2f b;
      b.x = w1a[t * 16 + mloc];
      b.y = w1b[t * 16 + mloc];
      acc[t] = __builtin_amdgcn_wmma_f32_16x16x4_f32(
          false, a, false, b, (short)0, acc[t], false, false);
    }
  }
  // tail: k0 = 1432, only k=1432 (lanes 0-15, first element) is valid
  {
    const int ka = (IN_DIM - 1) + khalf;
    const bool ok = (ka < IN_DIM);
    v2f a;
    a.x = ok ? xrow[ka] : 0.0f;
    a.y = 0.0f;
#pragma unroll
    for (int t = 0; t < 4; ++t) {
      v2f b;
      b.x = ok ? w1[(size_t)ka * HID + t * 16 + mloc] : 0.0f;
      b.y = 0.0f;
      acc[t] = __builtin_amdgcn_wmma_f32_16x16x4_f32(
          false, a, false, b, (short)0, acc[t], false, false);
    }
  }

  // unpack C fragments: VGPR g, lanes 0-15 -> M=g, lanes 16-31 -> M=g+8
  const int mrow = (lane >> 4) * 8;
#pragma unroll
  for (int t = 0; t < 4; ++t) {
    const int n = t * 16 + mloc;
    const float bias = b1[n];
#pragma unroll
    for (int g = 0; g < 8; ++g) {
      float v = acc[t][g] + bias;
      hsm[g + mrow][n] = v > 0.0f ? v : 0.0f;   // relu
    }
  }
  __syncthreads();

  // second layer: 16x7 outputs from 16x64 hidden tile
  for (int idx = lane; idx < 16 * OUT_C; idx += 32) {
    const int m = idx / OUT_C;
    const int c = idx % OUT_C;
    float s = b2[c];
#pragma unroll 8
    for (int n = 0; n < HID; ++n) s += hsm[m][n] * w2[n * OUT_C + c];
    F0[(size_t)(m0 + m) * OUT_C + c] = s;
  }
}

// ---------------------------------------------------------------------------
// Kernel 2: per-row degree (A + I) and deterministic CSR neighbor list.
// One 256-thread block per row; thread t owns columns [t*16, t*16+16).
// Count -> LDS inclusive scan -> ordered compaction (fixed edge order).
// ---------------------------------------------------------------------------
__global__ __launch_bounds__(256) void build_csr(
    const float* __restrict__ A, float* __restrict__ D,
    float* __restrict__ rsqD, int* __restrict__ nbr, int* __restrict__ nbr_cnt)
{
  __shared__ int cnt[256];
  const int i = blockIdx.x;
  const int t = threadIdx.x;
  const float* Arow = A + (size_t)i * N_NODES;
  const int j0 = t * 16;

  __builtin_prefetch(Arow + j0, 0, 0);

  int c = 0;
#pragma unroll
  for (int jj = 0; jj < 16; ++jj) c += (Arow[j0 + jj] != 0.0f) ? 1 : 0;
  cnt[t] = c;
  __syncthreads();

  // Hillis-Steele inclusive scan over 256 counts
  for (int off = 1; off < 256; off <<= 1) {
    int v = (t >= off) ? cnt[t - off] : 0;
    __syncthreads();
    cnt[t] += v;
    __syncthreads();
  }
  const int total  = cnt[255];
  const int offset = cnt[t] - c;   // exclusive prefix for this thread

  // compaction pass (sorted neighbor order -> deterministic FP sums later)
  int pos = offset;
#pragma unroll
  for (int jj = 0; jj < 16; ++jj) {
    if (Arow[j0 + jj] != 0.0f) {
      if (pos < MAX_DEG) nbr[(size_t)i * MAX_DEG + pos] = j0 + jj;
      ++pos;
    }
  }
  if (t == 0) {
    const float d = (float)total + 1.0f;   // + self loop
    D[i] = d;
    rsqD[i] = rsqrtf(d);
    nbr_cnt[i] = total < MAX_DEG ? total : MAX_DEG;
  }
}

// ---------------------------------------------------------------------------
// Kernel 3: one RUNG smoothing step. One wave per row. Edges only (W is
// always masked by A; diagonal of W is zeroed, so self loop contributes 0).
// ---------------------------------------------------------------------------
__global__ __launch_bounds__(32) void rung_step(
    const int* __restrict__ nbr, const int* __restrict__ nbr_cnt,
    const float* __restrict__ F0, const float* __restrict__ Fsrc,
    float* __restrict__ Fdst, const float* __restrict__ D,
    const float* __restrict__ rsqD, const float* __restrict__ log_gamma_0,
    const float* __restrict__ raw_decay, int kstep)
{
  __shared__ float red[32 * 8];
  const int i = blockIdx.x;
  const int t = threadIdx.x;

  const float g0  = expf(log_gamma_0[0]);
  const float r   = 1.0f / (1.0f + expf(-raw_decay[0]));
  const float lam = g0 * powf(r, (float)kstep) / SCAD_A;
  const float alam = SCAD_A * lam;

  __builtin_prefetch(F0 + (size_t)i * OUT_C, 0, 0);

  const float rsi = rsqD[i];
  float fni[OUT_C];
#pragma unroll
  for (int c = 0; c < OUT_C; ++c) fni[c] = Fsrc[(size_t)i * OUT_C + c] * rsi;

  const int deg = nbr_cnt[i];
  const int* row = nbr + (size_t)i * MAX_DEG;

  float qsum = 0.0f;
  float acc[OUT_C];
#pragma unroll
  for (int c = 0; c < OUT_C; ++c) acc[c] = 0.0f;

  for (int e = t; e < deg; e += 32) {
    const int j = row[e];
    const float rsj = rsqD[j];
    float fj[OUT_C];
    float d2 = 0.0f;
#pragma unroll
    for (int c = 0; c < OUT_C; ++c) {
      fj[c] = Fsrc[(size_t)j * OUT_C + c];
      const float dd = fni[c] - fj[c] * rsj;
      d2 += dd * dd;
    }
    const float y = sqrtf(d2);
    float w;
    if (y <= lam)       w = 1.0f;
    else if (y <= alam) w = (alam - y) / ((SCAD_A - 1.0f) * fmaxf(y, 1e-12f));
    else                w = 0.0f;
    if (w != w) w = 1.0f;            // NaN guard, as in reference

    qsum += w;                        // A entries are binary 1.0 on edges
    const float coef = w * rsi * rsj; // w * A_tilde_ij
#pragma unroll
    for (int c = 0; c < OUT_C; ++c) acc[c] += coef * fj[c];
  }

  red[t * 8 + 0] = qsum;
#pragma unroll
  for (int c = 0; c < OUT_C; ++c) red[t * 8 + 1 + c] = acc[c];
  __syncthreads();
  for (int s = 16; s > 0; s >>= 1) {
    if (t < s) {
#pragma unroll
      for (int u = 0; u < 8; ++u) red[t * 8 + u] += red[(t + s) * 8 + u];
    }
    __syncthreads();
  }

  if (t == 0) {
    const float Q = red[0] / D[i] + LAM_W;
    const float invQ = 1.0f / Q;
#pragma unroll
    for (int c = 0; c < OUT_C; ++c) {
      Fdst[(size_t)i * OUT_C + c] =
          red[1 + c] * invQ + LAM_W * F0[(size_t)i * OUT_C + c] * invQ;
    }
  }
}

// ---------------------------------------------------------------------------
extern "C" void kernel_launch(void* const* d_in, const int* in_sizes, int n_in,
                              void* d_out, int out_size, void* d_ws, size_t ws_size,
                              hipStream_t stream) {
  const float* A    = (const float*)d_in[0];
  const float* X    = (const float*)d_in[1];
  const float* w1   = (const float*)d_in[2];
  const float* b1   = (const float*)d_in[3];
  const float* w2   = (const float*)d_in[4];
  const float* b2   = (const float*)d_in[5];
  const float* lg0  = (const float*)d_in[6];
  const float* rdec = (const float*)d_in[7];
  float* out = (float*)d_out;

  // workspace carve-up (floats then ints; all 4-byte aligned)
  float* wsf  = (float*)d_ws;
  float* F0   = wsf;                               // 4096*7
  float* fb0  = F0  + (size_t)N_NODES * OUT_C;     // 4096*7
  float* fb1  = fb0 + (size_t)N_NODES * OUT_C;     // 4096*7
  float* D    = fb1 + (size_t)N_NODES * OUT_C;     // 4096
  float* rsqD = D   + N_NODES;                     // 4096
  int*   nbr  = (int*)(rsqD + N_NODES);            // 4096*256
  int*   ncnt = nbr + (size_t)N_NODES * MAX_DEG;   // 4096

  mlp_f0_wmma<<<dim3(N_NODES / 16), dim3(32), 0, stream>>>(X, w1, b1, w2, b2, F0);
  build_csr<<<dim3(N_NODES), dim3(256), 0, stream>>>(A, D, rsqD, nbr, ncnt);

  for (int k = 0; k < K_STEPS; ++k) {
    const float* src = (k == 0) ? F0 : ((k & 1) ? fb1 : fb0);
    float* dst = (k == K_STEPS - 1) ? out : ((k & 1) ? fb0 : fb1);
    rung_step<<<dim3(N_NODES), dim3(32), 0, stream>>>(
        nbr, ncnt, F0, src, dst, D, rsqD, lg0, rdec, k);
  }
}